// GenGradLoss_9706626089444
// MI455X (gfx1250) — compile-verified
//
#include <hip/hip_runtime.h>
#include <math.h>

// ---------------------------------------------------------------------------
// GenGradLoss for MI455X (gfx1250, wave32).
// Memory-bound: 2 x 75.5MB f32 streamed once (~6.5us floor at 23.3 TB/s).
// Neighbor rows (h+-1, d+-1) re-served from WGP$/192MB L2 (3-plane window
// = ~442KB), so DRAM traffic stays ~1x. Wave reductions use
// V_WMMA_F32_16X16X4_F32 (XDL) + one cross-half shuffle.
// ---------------------------------------------------------------------------

typedef __attribute__((ext_vector_type(2))) float v2f;
typedef __attribute__((ext_vector_type(8))) float v8f;

#define B_   4
#define D_   128
#define H_   192
#define W_   192
#define HW_  (H_ * W_)                 // 36864
#define N_   (B_ * D_ * H_ * W_)       // 18874368
#define BLK  256
#define NCHUNK   (N_ / 4)              // 4718592 float4-chunks
#define NBLOCKS  (NCHUNK / BLK)        // 18432 (exact)
#define NDSC 4608

// Wave32 sum via V_WMMA_F32_16X16X4_F32:
//   A: lane L supplies value at (M=L%16, K= L<16 ? 0 : 2); K=1,3 zeroed.
//   B: all ones  =>  D[m][n] = v_m + v_{m+16}  (every column identical).
//   D VGPR r holds rows M=r (lanes 0-15) and M=8+r (lanes 16-31); summing the
//   8 D registers per lane leaves two half-sums, combined with shfl_xor(16).
__device__ __forceinline__ float wave_sum_wmma(float v) {
  v2f a; a[0] = v;   a[1] = 0.0f;
  v2f b; b[0] = 1.0f; b[1] = 1.0f;
  v8f c = {};
  v8f d = __builtin_amdgcn_wmma_f32_16x16x4_f32(
      /*neg_a=*/false, a, /*neg_b=*/false, b,
      /*c_mod=*/(short)0, c, /*reuse_a=*/false, /*reuse_b=*/false);
  float s = ((d[0] + d[1]) + (d[2] + d[3])) + ((d[4] + d[5]) + (d[6] + d[7]));
  s += __shfl_xor(s, 16, 32);
  return s;  // full wave32 sum in every lane
}

struct Grad4 {
  float v[4];      // center values
  float g[3][4];   // |gradient| along W,H,D for the 4 elements
};

__device__ __forceinline__ Grad4 load_grads(const float* __restrict__ X,
                                            long base, long hm, long hp,
                                            long dm, long dp,
                                            int w, int h, int d) {
  Grad4 o;
  const float4 c   = *(const float4*)(X + base);
  const float4 xhm = *(const float4*)(X + base + hm);
  const float4 xhp = *(const float4*)(X + base + hp);
  const float4 xdm = *(const float4*)(X + base + dm);
  const float4 xdp = *(const float4*)(X + base + dp);
  const float wl = (w > 0)        ? X[base - 1] : c.x;   // value at w-1 (edge-clamped)
  const float wr = (w + 4 < W_)   ? X[base + 4] : c.w;   // value at w+4 (edge-clamped)

  const float cv[6] = { wl, c.x, c.y, c.z, c.w, wr };
  const float hsc = (h == 0 || h == H_ - 1) ? 1.0f : 0.5f;
  const float dsc = (d == 0 || d == D_ - 1) ? 1.0f : 0.5f;
  const float* hmv = (const float*)&xhm;
  const float* hpv = (const float*)&xhp;
  const float* dmv = (const float*)&xdm;
  const float* dpv = (const float*)&xdp;

#pragma unroll
  for (int i = 0; i < 4; ++i) {
    const int   wi  = w + i;
    const float wsc = (wi == 0 || wi == W_ - 1) ? 1.0f : 0.5f;
    o.v[i]    = cv[i + 1];
    o.g[0][i] = fabsf((cv[i + 2] - cv[i]) * wsc);      // d/dW
    o.g[1][i] = fabsf((hpv[i] - hmv[i]) * hsc);        // d/dH
    o.g[2][i] = fabsf((dpv[i] - dmv[i]) * dsc);        // d/dD
  }
  return o;
}

__global__ __launch_bounds__(BLK) void loss_main(const float* __restrict__ y,
                                                 const float* __restrict__ p,
                                                 float* __restrict__ partial) {
  const int t  = blockIdx.x * BLK + threadIdx.x;   // chunk id, exact coverage
  const int wc = t % (W_ / 4);
  int r        = t / (W_ / 4);
  const int h  = r % H_;  r /= H_;
  const int d  = r % D_;
  const int b  = r / D_;
  const int w  = wc * 4;
  const long base = ((long)(b * D_ + d) * H_ + h) * (long)W_ + w;

  const long hm = (h > 0)      ? -(long)W_  : 0;
  const long hp = (h < H_ - 1) ?  (long)W_  : 0;
  const long dm = (d > 0)      ? -(long)HW_ : 0;
  const long dp = (d < D_ - 1) ?  (long)HW_ : 0;

  // gfx1250 global_prefetch_b8: pull the streaming front 4 rows ahead.
  __builtin_prefetch(y + base + 4 * W_, 0, 0);
  __builtin_prefetch(p + base + 4 * W_, 0, 0);

  const Grad4 gy = load_grads(y, base, hm, hp, dm, dp, w, h, d);
  const Grad4 gp = load_grads(p, base, hm, hp, dm, dp, w, h, d);

  float l1 = 0.0f, gd = 0.0f;
#pragma unroll
  for (int i = 0; i < 4; ++i) {
    l1 += fabsf(gy.v[i] - gp.v[i]);
#pragma unroll
    for (int ax = 0; ax < 3; ++ax) {
      const float e = gy.g[ax][i] - gp.g[ax][i];
      gd = fmaf(e, e, gd);
    }
  }

  // Uniform control flow up to here: EXEC all-ones as WMMA requires.
  const float wl1 = wave_sum_wmma(l1);
  const float wgd = wave_sum_wmma(gd);

  __shared__ float red[16];
  const int lane = threadIdx.x & 31;
  const int wv   = threadIdx.x >> 5;
  if (lane == 0) { red[wv] = wl1; red[8 + wv] = wgd; }
  __syncthreads();
  if (threadIdx.x == 0) {
    float a = 0.0f, g = 0.0f;
#pragma unroll
    for (int i = 0; i < 8; ++i) { a += red[i]; g += red[8 + i]; }
    partial[2 * blockIdx.x + 0] = a;
    partial[2 * blockIdx.x + 1] = g;
  }
}

__global__ __launch_bounds__(BLK) void loss_final(const float* __restrict__ dsc,
                                                  const float* __restrict__ partial,
                                                  float* __restrict__ out) {
  float bce = 0.0f, l1 = 0.0f, gd = 0.0f;

  // BCE with logits vs zeros: 4608 = 18*256 -> divergence-free loop.
  for (int i = threadIdx.x; i < NDSC; i += BLK) {
    const float x = dsc[i];
    bce += fmaxf(x, 0.0f) + log1pf(expf(-fabsf(x)));
  }
  // Block partials: 18432 = 72*256 -> divergence-free loop.
  for (int i = threadIdx.x; i < NBLOCKS; i += BLK) {
    l1 += partial[2 * i + 0];
    gd += partial[2 * i + 1];
  }

  const float wb = wave_sum_wmma(bce);
  const float w1 = wave_sum_wmma(l1);
  const float wg = wave_sum_wmma(gd);

  __shared__ float red[24];
  const int lane = threadIdx.x & 31;
  const int wv   = threadIdx.x >> 5;
  if (lane == 0) { red[wv] = wb; red[8 + wv] = w1; red[16 + wv] = wg; }
  __syncthreads();
  if (threadIdx.x == 0) {
    float sb = 0.0f, sl = 0.0f, sg = 0.0f;
#pragma unroll
    for (int i = 0; i < 8; ++i) {
      sb += red[i]; sl += red[8 + i]; sg += red[16 + i];
    }
    out[0] = sb / (float)NDSC
           + 100.0f * (sl / (float)N_)
           + 100.0f * (sg / (float)N_);
  }
}

extern "C" void kernel_launch(void* const* d_in, const int* in_sizes, int n_in,
                              void* d_out, int out_size, void* d_ws, size_t ws_size,
                              hipStream_t stream) {
  const float* dsc_fake = (const float*)d_in[0];  // (4,1,8,12,12)
  const float* predicts = (const float*)d_in[1];  // (4,1,128,192,192)
  const float* y_data   = (const float*)d_in[2];  // (4,1,128,192,192)
  // d_in[3] (zeros) is unused: x * 0 == 0 in the stable BCE form.

  float* partial = (float*)d_ws;                  // 18432 * 2 floats = 147456 B

  loss_main <<<NBLOCKS, BLK, 0, stream>>>(y_data, predicts, partial);
  loss_final<<<1,       BLK, 0, stream>>>(dsc_fake, partial, (float*)d_out);
}